// GHP_31052613550343
// MI455X (gfx1250) — compile-verified
//
#include <hip/hip_runtime.h>
#include <math.h>

#define Cc   32
#define Kk   128
#define Bb   4
#define Tt   512
#define TPp  1024
#define CH   64
#define RT   2
#define EPS_F 2.2204460492503131e-16f

typedef __attribute__((ext_vector_type(2))) float v2f;
typedef __attribute__((ext_vector_type(8))) float v8f;

__device__ __forceinline__ float softplus_f(float x) {
    // matches jax.nn.softplus = max(x,0) + log1p(exp(-|x|))
    return fmaxf(x, 0.0f) + log1pf(__expf(-fabsf(x)));
}

// One-shot parameter prep: softplus tables, softplus(mu), fine probs, and the
// sparse-expansion matrix Mexp[c][k] = fine_prob[k] * (ftc[k]==c).
__global__ void prep_params(const float* __restrict__ mu,
                            const float* __restrict__ alpha,
                            const float* __restrict__ delta,
                            const float* __restrict__ cf,
                            const int*   __restrict__ ftc,
                            float* __restrict__ sp_alpha,
                            float* __restrict__ sp_delta,
                            float* __restrict__ sp_mu,
                            float* __restrict__ Mexp) {
    __shared__ float s_e[Kk];
    __shared__ float s_den[Cc];
    __shared__ float s_prob[Kk];
    __shared__ float s_max;
    const int tid = threadIdx.x;  // 128 threads

    for (int i = tid; i < Cc * Cc; i += 128) {
        sp_alpha[i] = softplus_f(alpha[i]);
        sp_delta[i] = softplus_f(delta[i]) + EPS_F;
    }
    if (tid < Cc) sp_mu[tid] = softplus_f(mu[tid]);

    if (tid == 0) {
        float m = cf[0];
        for (int k = 1; k < Kk; ++k) m = fmaxf(m, cf[k]);
        s_max = m;
    }
    __syncthreads();
    s_e[tid] = __expf(cf[tid] - s_max);
    __syncthreads();
    if (tid == 0) {
        for (int c = 0; c < Cc; ++c) s_den[c] = 0.0f;
        for (int k = 0; k < Kk; ++k) s_den[ftc[k]] += s_e[k];
    }
    __syncthreads();
    s_prob[tid] = s_e[tid] / s_den[ftc[tid]];
    __syncthreads();
    for (int i = tid; i < Cc * Kk; i += 128) {
        int c = i >> 7, k = i & (Kk - 1);
        Mexp[i] = (ftc[k] == c) ? s_prob[k] : 0.0f;
    }
}

// Main kernel: 256 blocks = B * (TP/16); 256 threads = 8 wave32.
// lane == coarse channel c; wave w owns query times t_base + 2w + {0,1};
// epilogue: wave w computes output k-tile w via 8 chained f32 WMMAs (K=32).
__global__ __launch_bounds__(256)
void hawkes_wmma_kernel(const int*   __restrict__ past_event,
                        const float* __restrict__ past_time,
                        const float* __restrict__ time_tensor,
                        const float* __restrict__ sp_alpha,
                        const float* __restrict__ sp_delta,
                        const float* __restrict__ sp_mu,
                        const float* __restrict__ Mexp,
                        float*       __restrict__ out) {
    __shared__ float s_spA[Cc * Cc];   // 4 KB
    __shared__ float s_spD[Cc * Cc];   // 4 KB
    __shared__ float s_mu[Cc];
    __shared__ float s_pt[CH];
    __shared__ int   s_ev[CH];
    __shared__ float s_acc[16 * Cc];   // 2 KB : [t_local][c]

    const int tid    = threadIdx.x;
    const int lane   = tid & 31;          // coarse channel c
    const int wave   = tid >> 5;          // 0..7
    const int b      = blockIdx.x >> 6;   // 64 time-tiles per batch
    const int tile   = blockIdx.x & 63;
    const int t_base = tile * 16;

    // Stage softplus tables into LDS (shared by all 8 waves).
    for (int i = tid; i < Cc * Cc; i += 256) {
        s_spA[i] = sp_alpha[i];
        s_spD[i] = sp_delta[i];
    }
    if (tid < Cc) s_mu[tid] = sp_mu[tid];
    __builtin_prefetch(Mexp + lane * 16, 0, 0);  // warm cache for the WMMA epilogue

    float tm[RT], acc[RT];
#pragma unroll
    for (int r = 0; r < RT; ++r) {
        const int t = t_base + wave * RT + r;
        tm[r]  = time_tensor[b * TPp + t];
        acc[r] = 0.0f;
    }

    // Accumulate Hawkes intensities over all T past events, chunked via LDS.
    for (int chunk = 0; chunk < Tt; chunk += CH) {
        __syncthreads();
        if (tid < CH) {
            s_pt[tid] = past_time[b * Tt + chunk + tid];
            s_ev[tid] = past_event[b * Tt + chunk + tid];
        }
        __syncthreads();
#pragma unroll 4
        for (int ii = 0; ii < CH; ++ii) {
            const float pt = s_pt[ii];
            const int   ev = s_ev[ii];
            const float a  = s_spA[ev * Cc + lane];
            const float d  = s_spD[ev * Cc + lane];
#pragma unroll
            for (int r = 0; r < RT; ++r) {
                const float et  = tm[r] - pt;
                const float etc = fmaxf(et, 0.0f);            // clamp so exp never overflows
                const float val = a * __expf(-d * etc);       // v_exp_f32 (TRANS)
                acc[r] += (et > EPS_F) ? val : 0.0f;          // causal mask
            }
        }
    }

    // Write (acc + softplus(mu)) into WMMA-friendly LDS tile [16 t][32 c].
    __syncthreads();
#pragma unroll
    for (int r = 0; r < RT; ++r) {
        s_acc[(wave * RT + r) * Cc + lane] = acc[r] + s_mu[lane];
    }
    __syncthreads();

    // Epilogue: D[16x16] = A[16x32] x Mexp[32x(16 cols of K)] with
    // 8 chained V_WMMA_F32_16X16X4_F32 (full EXEC, no divergence here).
    const int half = lane >> 4;   // K-half select per ISA A/B layout
    const int l15  = lane & 15;
    v8f cacc = {};
#pragma unroll
    for (int kc = 0; kc < 8; ++kc) {
        v2f af, bf;
#pragma unroll
        for (int j = 0; j < 2; ++j) {
            const int cch = kc * 4 + half * 2 + j;           // K index within 32
            af[j] = s_acc[l15 * Cc + cch];                   // A: M=l15, K=cch
            bf[j] = Mexp[cch * Kk + wave * 16 + l15];        // B: K=cch, N=l15 (tile=wave)
        }
        cacc = __builtin_amdgcn_wmma_f32_16x16x4_f32(
            /*neg_a=*/false, af, /*neg_b=*/false, bf,
            /*c_mod=*/(short)0, cacc, /*reuse_a=*/false, /*reuse_b=*/false);
    }
#pragma unroll
    for (int r = 0; r < 8; ++r) {
        const int m = r + half * 8;                          // D row per ISA layout
        const int t = t_base + m;
        out[(b * TPp + t) * Kk + wave * 16 + l15] = cacc[r];
    }
}

extern "C" void kernel_launch(void* const* d_in, const int* in_sizes, int n_in,
                              void* d_out, int out_size, void* d_ws, size_t ws_size,
                              hipStream_t stream) {
    const int*   past_event  = (const int*)  d_in[0];
    const float* past_time   = (const float*)d_in[1];
    const float* time_tensor = (const float*)d_in[2];
    const float* mu          = (const float*)d_in[3];
    const float* alpha       = (const float*)d_in[4];
    const float* delta       = (const float*)d_in[5];
    const float* cf_logits   = (const float*)d_in[6];
    const int*   ftc         = (const int*)  d_in[7];
    float* out = (float*)d_out;

    float* sp_alpha = (float*)d_ws;          // 1024 f32
    float* sp_delta = sp_alpha + Cc * Cc;    // 1024 f32
    float* sp_mu    = sp_delta + Cc * Cc;    // 32 f32
    float* Mexp     = sp_mu + Cc;            // 4096 f32  (~25 KB total)

    prep_params<<<1, 128, 0, stream>>>(mu, alpha, delta, cf_logits, ftc,
                                       sp_alpha, sp_delta, sp_mu, Mexp);
    hawkes_wmma_kernel<<<Bb * (TPp / 16), 256, 0, stream>>>(
        past_event, past_time, time_tensor,
        sp_alpha, sp_delta, sp_mu, Mexp, out);
}